// GNN_57870389346990
// MI455X (gfx1250) — compile-verified
//
#include <hip/hip_runtime.h>
#include <hip/hip_bf16.h>

// Problem constants from the reference setup.
#define NB 8       // batches
#define NN 2000    // nodes
#define DD 64      // embedding dim
#define RR 16      // edge-feature rank
#define NEG_SLOPE 0.2f

typedef float v2f __attribute__((ext_vector_type(2)));
typedef float v8f __attribute__((ext_vector_type(8)));

// ---------------------------------------------------------------------------
// Kernel 1: stream A (256 MB) once. Per edge (i,j):
//   G[i,j] = leaky_relu(A[i,j,:]·W + b) * valid(i,j)
// and accumulate deg[i] = sum_j valid(i,j) via wave reduce + atomicAdd.
// ---------------------------------------------------------------------------
__global__ void gnn_gate_kernel(const float* __restrict__ A,
                                const float* __restrict__ W,
                                const float* __restrict__ bptr,
                                float* __restrict__ G,
                                float* __restrict__ deg) {
    const int i = blockIdx.y;
    const int j = blockIdx.x * blockDim.x + threadIdx.x;

    float valid = 0.0f;
    if (j < NN) {
        const float* a = A + ((size_t)i * NN + j) * RR;
        float s = 0.0f, t = 0.0f;
#pragma unroll
        for (int r = 0; r < RR; r += 4) {
            float4 av = *(const float4*)(a + r);
            float4 wv = *(const float4*)(W + r);
            s += av.x * wv.x + av.y * wv.y + av.z * wv.z + av.w * wv.w;
            t += av.x + av.y + av.z + av.w;
        }
        float g = s + bptr[0];
        g = (g > 0.0f) ? g : NEG_SLOPE * g;
        valid = (t > 0.0f) ? 1.0f : 0.0f;
        G[(size_t)i * NN + j] = g * valid;
    }
    // wave32 reduction of valid, one atomic per wave
    float v = valid;
#pragma unroll
    for (int off = 16; off > 0; off >>= 1)
        v += __shfl_down(v, off, 32);
    if ((threadIdx.x & 31) == 0 && v != 0.0f)
        atomicAdd(&deg[i], v);
}

// ---------------------------------------------------------------------------
// Kernel 2: fused  e_new[b] = ((E Eᵀ) ∘ G) E / deg  +  concat with e_old.
// Grid: 125 I-tiles; block: 256 threads = 8 waves, wave w handles batch w.
// Per J-tile:  S(16x16) = E_I · E_Jᵀ   (16× v_wmma_f32_16x16x4_f32)
//              S *= G tile             (VALU, C/D layout)
//              S: D-layout -> A-layout via padded LDS roundtrip
//              acc(16x64) += S · E_J   (16× v_wmma_f32_16x16x4_f32)
// ---------------------------------------------------------------------------
__global__ void __launch_bounds__(256)
gnn_fused_kernel(const float* __restrict__ E,      // [B,N,D]
                 const float* __restrict__ G,      // [N,N]
                 const float* __restrict__ deg,    // [N]
                 float* __restrict__ out) {        // [B,N,2D]
    __shared__ float s_stage[NB][16 * 18];         // stride 18: bank-clean + 8B-aligned pairs

    const int wave = threadIdx.x >> 5;             // 0..7 == batch index
    const int lane = threadIdx.x & 31;
    const int m    = lane & 15;                    // row/col within tile
    const int h    = lane >> 4;                    // lane half
    const int b    = wave;
    const int I    = blockIdx.x * 16;

    const float* Eb = E + (size_t)b * NN * DD;
    float* slds = s_stage[wave];

    // Preload A-fragments of E_I (16x64): lane l holds E[I+m, 4t+2h .. +1]
    v2f afrag[16];
#pragma unroll
    for (int t = 0; t < 16; ++t)
        afrag[t] = *(const v2f*)(Eb + (size_t)(I + m) * DD + 4 * t + 2 * h);

    v8f acc[4] = {v8f{0}, v8f{0}, v8f{0}, v8f{0}};   // 16 rows x 64 cols accumulator

    for (int J = 0; J < NN; J += 16) {
        // ---- S = E_I * E_J^T  (K = 64 in 16 steps of 4) ----
        v8f S = {};
#pragma unroll
        for (int t = 0; t < 16; ++t) {
            v2f bfrag = *(const v2f*)(Eb + (size_t)(J + m) * DD + 4 * t + 2 * h);
            S = __builtin_amdgcn_wmma_f32_16x16x4_f32(
                    false, afrag[t], false, bfrag, (short)0, S, false, false);
        }

        // ---- S *= G tile (C/D layout: element (M=r+8h, N=m)) ----
#pragma unroll
        for (int r = 0; r < 8; ++r)
            S[r] *= G[(size_t)(I + r + 8 * h) * NN + (J + m)];

        // ---- D-layout -> A-layout through LDS ----
#pragma unroll
        for (int r = 0; r < 8; ++r)
            slds[(r + 8 * h) * 18 + m] = S[r];
        asm volatile("s_wait_dscnt 0" ::: "memory");   // cross-lane visibility within wave
        v2f sA[4];
#pragma unroll
        for (int t = 0; t < 4; ++t)
            sA[t] = *(const v2f*)(slds + m * 18 + 4 * t + 2 * h);

        // ---- acc += S * E_J  (K = 16, 4 column chunks of 16) ----
#pragma unroll
        for (int c = 0; c < 4; ++c) {
#pragma unroll
            for (int t = 0; t < 4; ++t) {
                v2f bfrag;
                bfrag.x = Eb[(size_t)(J + 4 * t + 2 * h) * DD + c * 16 + m];
                bfrag.y = Eb[(size_t)(J + 4 * t + 2 * h + 1) * DD + c * 16 + m];
                acc[c] = __builtin_amdgcn_wmma_f32_16x16x4_f32(
                             false, sA[t], false, bfrag, (short)0, acc[c], false, false);
            }
        }
        asm volatile("s_wait_dscnt 0" ::: "memory");   // reads done before next iter's stores
    }

    // ---- epilogue: out[b, row, 64:128] = acc / deg[row] ----
#pragma unroll
    for (int r = 0; r < 8; ++r) {
        const int row = I + r + 8 * h;
        const float inv_d = 1.0f / fmaxf(deg[row], 1.0f);
        float* orow = out + ((size_t)b * NN + row) * (2 * DD);
#pragma unroll
        for (int c = 0; c < 4; ++c)
            orow[DD + c * 16 + m] = acc[c][r] * inv_d;
    }

    // ---- epilogue: out[b, row, 0:64] = e_old (lane copies 32 floats) ----
    {
        const int row = I + m;
        const float* erow = Eb + (size_t)row * DD + 32 * h;
        float* orow = out + ((size_t)b * NN + row) * (2 * DD) + 32 * h;
#pragma unroll
        for (int q = 0; q < 8; ++q)
            *(float4*)(orow + 4 * q) = *(const float4*)(erow + 4 * q);
    }
}

// ---------------------------------------------------------------------------
// Launch
// ---------------------------------------------------------------------------
extern "C" void kernel_launch(void* const* d_in, const int* in_sizes, int n_in,
                              void* d_out, int out_size, void* d_ws, size_t ws_size,
                              hipStream_t stream) {
    const float* e_old = (const float*)d_in[0];   // [B,N,D]
    const float* A     = (const float*)d_in[1];   // [N,N,R]
    const float* W     = (const float*)d_in[2];   // [R,1]
    const float* bias  = (const float*)d_in[3];   // [1]
    float* out = (float*)d_out;                   // [B,N,2D]

    float* deg = (float*)d_ws;                    // N floats (padded to 2048)
    float* G   = deg + 2048;                      // N*N floats (16 MB)

    hipMemsetAsync(deg, 0, NN * sizeof(float), stream);

    dim3 gate_grid((NN + 255) / 256, NN);
    gnn_gate_kernel<<<gate_grid, 256, 0, stream>>>(A, W, bias, G, deg);

    gnn_fused_kernel<<<NN / 16, 256, 0, stream>>>(e_old, G, deg, out);
}